// ExampleModel_1116691497724
// MI455X (gfx1250) — compile-verified
//
#include <hip/hip_runtime.h>

// Problem dims (fixed by the reference)
#define B_ 4
#define T_ 512
#define D_ 2048
#define H_ 1024
#define E_ 2

typedef __attribute__((ext_vector_type(2))) float v2f;
typedef __attribute__((ext_vector_type(8))) float v8f;

// ---------------------------------------------------------------------------
// Block-wide sum reduction through LDS (blockDim.x == 256, 8 wave32s).
// ---------------------------------------------------------------------------
__device__ __forceinline__ float block_reduce_sum(float v, float* sm) {
    int tid = threadIdx.x;
    sm[tid] = v;
    __syncthreads();
    for (int off = 128; off > 0; off >>= 1) {
        if (tid < off) sm[tid] += sm[tid + off];
        __syncthreads();
    }
    float r = sm[0];
    __syncthreads();
    return r;
}

// ---------------------------------------------------------------------------
// w2sum[r] = sum_d W2[r*D + d], r in [0, E*H). One block per row.
// ---------------------------------------------------------------------------
__global__ void k_w2sum(const float* __restrict__ W2, float* __restrict__ w2sum) {
    __shared__ float sm[256];
    int r = blockIdx.x;
    const float* row = W2 + (size_t)r * D_;
    float p = 0.f;
    for (int d = threadIdx.x; d < D_; d += 256) p += row[d];
    float s = block_reduce_sum(p, sm);
    if (threadIdx.x == 0) w2sum[r] = s;
}

// ---------------------------------------------------------------------------
// v[r] = sum_h W1[r*H + h] * w2sum[(r/D)*H + h], r in [0, E*D).
// ---------------------------------------------------------------------------
__global__ void k_vvec(const float* __restrict__ W1, const float* __restrict__ w2sum,
                       float* __restrict__ v) {
    __shared__ float sm[256];
    int r = blockIdx.x;
    int e = r / D_;
    const float* row = W1 + (size_t)r * H_;
    const float* ws  = w2sum + (size_t)e * H_;
    float p = 0.f;
    for (int h = threadIdx.x; h < H_; h += 256) p += row[h] * ws[h];
    float s = block_reduce_sum(p, sm);
    if (threadIdx.x == 0) v[r] = s;
}

// ---------------------------------------------------------------------------
// cvec[e] = dot(b1[e], w2sum[e]) + sum_d b2[e,d]. One block per expert.
// ---------------------------------------------------------------------------
__global__ void k_cvec(const float* __restrict__ b1, const float* __restrict__ b2,
                       const float* __restrict__ w2sum, float* __restrict__ cvec) {
    __shared__ float sm[256];
    int e = blockIdx.x;
    float p = 0.f;
    for (int h = threadIdx.x; h < H_; h += 256) p += b1[e * H_ + h] * w2sum[e * H_ + h];
    for (int d = threadIdx.x; d < D_; d += 256) p += b2[e * D_ + d];
    float s = block_reduce_sum(p, sm);
    if (threadIdx.x == 0) cvec[e] = s;
}

// ---------------------------------------------------------------------------
// Pack the WMMA B operand once, in its exact register-fragment layout.
// Padded matrix V[D x 16]: col 0 -> Wg[:,0], col 1 -> Wg[:,1],
//                          col 2 -> v[0,:],  col 3 -> v[1,:], cols 4..15 -> 0.
// Fragment layout for V_WMMA_F32_16X16X4_F32 B (4x16), K-chunk c (k0 = 4c):
//   lane l holds (V[4c + 2*(l/16) + 0][l%16], V[4c + 2*(l/16) + 1][l%16])
// Bfrag is [D/4][32] float2 = 128 KB; one thread per (chunk, lane).
// ---------------------------------------------------------------------------
__global__ void k_bpack(const float* __restrict__ Wg, const float* __restrict__ v,
                        float* __restrict__ Bfrag) {
    int idx = blockIdx.x * 256 + threadIdx.x;        // 0 .. (D/4)*32 - 1
    int l = idx & 31;
    int c = idx >> 5;
    int n  = l & 15;
    int kb = (l >> 4) << 1;
    int k  = c * 4 + kb;
    float b0 = 0.f, b1v = 0.f;
    if      (n == 0) { b0 = Wg[2 * k];         b1v = Wg[2 * (k + 1)];     }
    else if (n == 1) { b0 = Wg[2 * k + 1];     b1v = Wg[2 * (k + 1) + 1]; }
    else if (n == 2) { b0 = v[k];              b1v = v[k + 1];            }
    else if (n == 3) { b0 = v[D_ + k];         b1v = v[D_ + k + 1];       }
    Bfrag[(size_t)idx * 2 + 0] = b0;
    Bfrag[(size_t)idx * 2 + 1] = b1v;
}

// ---------------------------------------------------------------------------
// Token stage: S = X[2048 x 2048] * V[2048 x 16] via V_WMMA_F32_16X16X4_F32.
// One wave32 per 16-token tile; 4 waves (128 threads) per workgroup.
// Hot loop: 1 coalesced b64 load (A), 1 coalesced b64 load (B), 1 WMMA.
// After the K-loop the C fragment is spilled through LDS and lanes 0..15 do
// the softmax gate / top-1 select / combine, producing s[b,t].
// ---------------------------------------------------------------------------
__global__ void __launch_bounds__(128) k_tokens(const float* __restrict__ x,
                                                const float* __restrict__ Bfrag,
                                                const float* __restrict__ cvec,
                                                float* __restrict__ s_out) {
    __shared__ float Ct[4][16][16];

    int wave = threadIdx.x >> 5;
    int lane = threadIdx.x & 31;
    int tile = blockIdx.x * 4 + wave;      // 16-token tile id, 0..127
    int m    = lane & 15;                  // A row (token within tile)
    int kb   = (lane >> 4) << 1;           // K sub-offset: 0 or 2

    // A side: lane streams its own token row (8-byte aligned: row start and kb even).
    const float* xp = x + ((size_t)tile * 16 + m) * D_ + kb;
    // B side: pre-swizzled fragment stream, consecutive lanes -> consecutive 8B.
    const v2f* bp = (const v2f*)Bfrag + lane;

    v8f acc = {};
#pragma unroll 4
    for (int c = 0; c < D_ / 4; ++c) {
        v2f a = *(const v2f*)(xp + 4 * c);
        v2f b = bp[c * 32];
        // D = A(16x4 f32) * B(4x16 f32) + C(16x16 f32)
        acc = __builtin_amdgcn_wmma_f32_16x16x4_f32(
            /*neg_a=*/false, a, /*neg_b=*/false, b,
            /*c_mod=*/(short)0, acc, /*reuse_a=*/false, /*reuse_b=*/false);
    }

    // Spill C fragment: VGPR j, lane l -> C[M][N], M = j + 8*(l/16), N = l%16.
#pragma unroll
    for (int j = 0; j < 8; ++j) {
        int M = j + ((lane >> 4) << 3);
        Ct[wave][M][m] = acc[j];
    }
    __syncthreads();

    // Lanes 0..15: one token each. Gate = softmax over 2 logits; top-1 combine.
    if (lane < 16) {
        float l0 = Ct[wave][lane][0];
        float l1 = Ct[wave][lane][1];
        float mx = fmaxf(l0, l1);
        float e0 = expf(l0 - mx), e1 = expf(l1 - mx);
        float inv = 1.f / (e0 + e1);
        float p0 = e0 * inv, p1 = e1 * inv;
        int   e    = (p1 > p0) ? 1 : 0;            // argmax, first index on tie
        float gate = e ? p1 : p0;
        float dot  = Ct[wave][lane][2 + e];        // x . v[e]
        s_out[tile * 16 + lane] = gate * (dot + cvec[e]);
    }
}

// ---------------------------------------------------------------------------
// Per-batch log_softmax over T=512: out[b,t] = s[b,t] - (max + log(sum exp)).
// ---------------------------------------------------------------------------
__global__ void k_logsoftmax(const float* __restrict__ s, float* __restrict__ out) {
    __shared__ float sm[256];
    int b = blockIdx.x;
    int tid = threadIdx.x;
    const float* row = s + (size_t)b * T_;

    float m = -3.402823466e38f;
    for (int t = tid; t < T_; t += 256) m = fmaxf(m, row[t]);
    sm[tid] = m;
    __syncthreads();
    for (int off = 128; off > 0; off >>= 1) {
        if (tid < off) sm[tid] = fmaxf(sm[tid], sm[tid + off]);
        __syncthreads();
    }
    float mx = sm[0];
    __syncthreads();

    float p = 0.f;
    for (int t = tid; t < T_; t += 256) p += expf(row[t] - mx);
    float tot = block_reduce_sum(p, sm);
    float lse = mx + logf(tot);

    for (int t = tid; t < T_; t += 256) out[(size_t)b * T_ + t] = row[t] - lse;
}

// ---------------------------------------------------------------------------
// Entry point
// ---------------------------------------------------------------------------
extern "C" void kernel_launch(void* const* d_in, const int* in_sizes, int n_in,
                              void* d_out, int out_size, void* d_ws, size_t ws_size,
                              hipStream_t stream) {
    (void)in_sizes; (void)n_in; (void)out_size; (void)ws_size;

    const float* x  = (const float*)d_in[0];   // [B,T,D]
    const float* Wg = (const float*)d_in[1];   // [D,E]
    const float* W1 = (const float*)d_in[2];   // [E,D,H]
    const float* b1 = (const float*)d_in[3];   // [E,H]
    const float* W2 = (const float*)d_in[4];   // [E,H,D]
    const float* b2 = (const float*)d_in[5];   // [E,D]
    float* out = (float*)d_out;                // [B,T]

    float* ws     = (float*)d_ws;
    float* w2sum  = ws;                        // E*H   = 2048 floats
    float* vvec   = w2sum + E_ * H_;           // E*D   = 4096 floats
    float* cvec   = vvec + E_ * D_;            // E (padded to 16)
    float* s      = cvec + 16;                 // B*T   = 2048 floats
    float* Bfrag  = s + B_ * T_;               // (D/4)*32*2 = 32768 floats (128 KB)

    k_w2sum     <<<E_ * H_,           256, 0, stream>>>(W2, w2sum);
    k_vvec      <<<E_ * D_,           256, 0, stream>>>(W1, w2sum, vvec);
    k_cvec      <<<E_,                256, 0, stream>>>(b1, b2, w2sum, cvec);
    k_bpack     <<<(D_ / 4) * 32/256, 256, 0, stream>>>(Wg, vvec, Bfrag);
    k_tokens    <<<(B_ * T_) / 64,    128, 0, stream>>>(x, Bfrag, cvec, s);
    k_logsoftmax<<<B_,                256, 0, stream>>>(s, out);
}